// SSCDNonLModel_gat_62594853372074
// MI455X (gfx1250) — compile-verified
//
#include <hip/hip_runtime.h>
#include <hip/hip_bf16.h>
#include <math.h>

// ---------------------------------------------------------------------------
// SSCDNonLModel_gat forward for MI455X (gfx1250).
// 1x1 convs run on V_WMMA_F32_16X16X4_F32. BatchNorm is folded into the
// *weights* (A' = A*s, bias' = bias + A@t) so the WMMA inner loop is fully
// branch-free: 2x b64 A' loads + 4x b32 B loads + 2 WMMAs per 8 K-steps.
// ---------------------------------------------------------------------------

typedef __attribute__((ext_vector_type(2))) float v2f;
typedef __attribute__((ext_vector_type(8))) float v8f;

#define NPIX 4096

__device__ __forceinline__ float sigmoidf_(float x) { return 1.0f / (1.0f + __expf(-x)); }
__device__ __forceinline__ float lrelu02(float x) { return x >= 0.0f ? x : 0.2f * x; }

// ---------------------------------------------------------------------------
// BatchNorm (training mode) stats over one channel row of NPIX values,
// folded into scale s[c] and shift t[c]:  bn(x) = x*s + t.
// ---------------------------------------------------------------------------
__global__ void bn_stats_fold(const float* __restrict__ x, const float* __restrict__ g,
                              const float* __restrict__ bt, float* __restrict__ s,
                              float* __restrict__ t) {
  __shared__ float rs[256], rq[256];
  const int c = blockIdx.x;
  const float* row = x + (size_t)c * NPIX;
  float ls = 0.f, lq = 0.f;
  for (int p = threadIdx.x; p < NPIX; p += 256) { float v = row[p]; ls += v; lq += v * v; }
  rs[threadIdx.x] = ls; rq[threadIdx.x] = lq; __syncthreads();
  for (int st = 128; st > 0; st >>= 1) {
    if (threadIdx.x < st) { rs[threadIdx.x] += rs[threadIdx.x + st]; rq[threadIdx.x] += rq[threadIdx.x + st]; }
    __syncthreads();
  }
  if (threadIdx.x == 0) {
    float mean = rs[0] * (1.0f / NPIX);
    float var  = rq[0] * (1.0f / NPIX) - mean * mean;
    float sc   = g[c] * rsqrtf(var + 1e-5f);
    s[c] = sc; t[c] = bt[c] - mean * sc;
  }
}

// ---------------------------------------------------------------------------
// Fold BN scale/shift into weights:
//   Ap[m,k]  = A[m,k] * s[k]          (zero-padded to [Mpad, Kpad])
//   biasp[m] = bias[m] + sum_k A[m,k] * t[k]
// s/t may be null (identity). grid.x = Mpad, block = 256.
// ---------------------------------------------------------------------------
__global__ void fold_weights(const float* __restrict__ A, const float* __restrict__ bias,
                             const float* __restrict__ s, const float* __restrict__ t,
                             float* __restrict__ Ap, float* __restrict__ biasp,
                             int M, int K, int Kpad) {
  __shared__ float red[256];
  const int m = blockIdx.x;
  float lb = 0.f;
  for (int k = threadIdx.x; k < Kpad; k += 256) {
    const bool ok = (m < M) && (k < K);
    const float a  = ok ? A[(size_t)m * K + k] : 0.f;
    const float sv = (k < K) ? (s ? s[k] : 1.f) : 0.f;
    const float tv = (k < K) ? (t ? t[k] : 0.f) : 0.f;
    Ap[(size_t)m * Kpad + k] = a * sv;
    lb += a * tv;
  }
  red[threadIdx.x] = lb; __syncthreads();
  for (int st = 128; st > 0; st >>= 1) {
    if (threadIdx.x < st) red[threadIdx.x] += red[threadIdx.x + st];
    __syncthreads();
  }
  if (threadIdx.x == 0) biasp[m] = (m < M ? bias[m] : 0.f) + red[0];
}

// ---------------------------------------------------------------------------
// C[M,N] = act( Ap[Mpad,Kpad] @ B[Kpad,N] + biasp[m] )
// Branch-free inner loop; one 16x16 tile per wave, 2 WMMAs (K=8) per iter
// with independent accumulators. block = 128 (4 waves); grid = (Mpad/16, N/64).
// Requires: Kpad % 8 == 0; B readable for Kpad rows (pad rows zeroed).
// f32 WMMA frags: A 16x4 (2 VGPR): lane<16 -> K={k,k+1}, lane>=16 -> K={k+2,k+3},
// M=lane%16. B 4x16: same K split, N=lane%16. C/D: M = v + 8*(lane>=16), N=lane%16.
// ---------------------------------------------------------------------------
__global__ void gemm_wmma_f32(const float* __restrict__ Ap, const float* __restrict__ B,
                              const float* __restrict__ biasp, float* __restrict__ C,
                              int M, int Kpad, int N, int act) {
  const int lane = threadIdx.x & 31;
  const int wave = threadIdx.x >> 5;
  const int m0 = blockIdx.x * 16;
  const int n0 = (blockIdx.y * 4 + wave) * 16;
  const int lm = lane & 15;
  const int hi = lane >> 4;
  const int ncol = n0 + lm;

  const float* ap = Ap + (size_t)(m0 + lm) * Kpad + 2 * hi;
  const float* bp = B + (size_t)(2 * hi) * N + ncol;

  v8f acc0 = {0.f, 0.f, 0.f, 0.f, 0.f, 0.f, 0.f, 0.f};
  v8f acc1 = {0.f, 0.f, 0.f, 0.f, 0.f, 0.f, 0.f, 0.f};
  for (int k0 = 0; k0 < Kpad; k0 += 8) {
    const v2f a0 = *(const v2f*)(ap);
    const v2f a1 = *(const v2f*)(ap + 4);
    v2f b0, b1;
    b0.x = bp[0];
    b0.y = bp[(size_t)N];
    b1.x = bp[(size_t)4 * N];
    b1.y = bp[(size_t)5 * N];
    acc0 = __builtin_amdgcn_wmma_f32_16x16x4_f32(false, a0, false, b0, (short)0, acc0, false, false);
    acc1 = __builtin_amdgcn_wmma_f32_16x16x4_f32(false, a1, false, b1, (short)0, acc1, false, false);
    ap += 8;
    bp += (size_t)8 * N;
  }
#pragma unroll
  for (int v = 0; v < 8; ++v) {
    const int m = m0 + v + 8 * hi;
    if (m < M) {
      float r = acc0[v] + acc1[v] + biasp[m];
      if (act) r = sigmoidf_(r);
      C[(size_t)m * N + ncol] = r;
    }
  }
}

// ---------------------------------------------------------------------------
// Grouped 5x5 conv, pad 2, + sigmoid. Output channel o uses input channels
// [o*cpg, o*cpg+cpg). Depthwise == cpg=1. in laid out [Cin, 64, 64].
// ---------------------------------------------------------------------------
__global__ void grouped_conv5x5_sig(const float* __restrict__ in, const float* __restrict__ w,
                                    const float* __restrict__ b, float* __restrict__ out,
                                    int Cout, int cpg) {
  const int idx = blockIdx.x * blockDim.x + threadIdx.x;
  if (idx >= Cout * NPIX) return;
  const int o = idx >> 12;
  const int p = idx & 4095;
  const int y = p >> 6, x = p & 63;
  float acc = b[o];
  for (int ci = 0; ci < cpg; ++ci) {
    const float* ip = in + (size_t)(o * cpg + ci) * NPIX;
    const float* wp = w + (size_t)(o * cpg + ci) * 25;
#pragma unroll
    for (int ky = 0; ky < 5; ++ky) {
      const int iy = y + ky - 2;
      if (iy < 0 || iy > 63) continue;
#pragma unroll
      for (int kx = 0; kx < 5; ++kx) {
        const int ix = x + kx - 2;
        if (ix < 0 || ix > 63) continue;
        acc += ip[iy * 64 + ix] * wp[ky * 5 + kx];
      }
    }
  }
  out[idx] = sigmoidf_(acc);
}

// out = in*s[c] + t[c]  (materialize bn4 output xx)
__global__ void bn_apply(const float* __restrict__ in, const float* __restrict__ s,
                         const float* __restrict__ t, float* __restrict__ out, int total) {
  const int idx = blockIdx.x * blockDim.x + threadIdx.x;
  if (idx >= total) return;
  const int c = idx >> 12;
  out[idx] = in[idx] * s[c] + t[c];
}

// ---------------------------------------------------------------------------
// cm = w_mask . xx + b_mask per pixel; softmax over all 4096 pixels;
// mask[p] = (softmax >= 0.0009). Also nodes = w_trans . xx + b_trans.
// Single block of 1024 threads.
// ---------------------------------------------------------------------------
__global__ void att_mask_nodes(const float* __restrict__ xx, const float* __restrict__ wm,
                               const float* __restrict__ bm, const float* __restrict__ wt,
                               const float* __restrict__ btr, float* __restrict__ mask,
                               float* __restrict__ nodes) {
  __shared__ float sh[NPIX];
  __shared__ float red[1024];
  const int tid = threadIdx.x;
  for (int p = tid; p < NPIX; p += 1024) {
    float cm = bm[0], nd = btr[0];
    for (int c = 0; c < 25; ++c) { float v = xx[c * NPIX + p]; cm += wm[c] * v; nd += wt[c] * v; }
    sh[p] = cm; nodes[p] = nd;
  }
  __syncthreads();
  float lm = -1e30f;
  for (int p = tid; p < NPIX; p += 1024) lm = fmaxf(lm, sh[p]);
  red[tid] = lm; __syncthreads();
  for (int st = 512; st > 0; st >>= 1) { if (tid < st) red[tid] = fmaxf(red[tid], red[tid + st]); __syncthreads(); }
  const float mx = red[0]; __syncthreads();
  float lsum = 0.f;
  for (int p = tid; p < NPIX; p += 1024) { float e = __expf(sh[p] - mx); sh[p] = e; lsum += e; }
  red[tid] = lsum; __syncthreads();
  for (int st = 512; st > 0; st >>= 1) { if (tid < st) red[tid] += red[tid + st]; __syncthreads(); }
  const float inv = 1.0f / red[0];
  __syncthreads();
  for (int p = tid; p < NPIX; p += 1024) mask[p] = (sh[p] * inv >= 0.0009f) ? 1.0f : 0.0f;
}

// ---------------------------------------------------------------------------
// GAT layer 1 scores: input feature dim == 1 => z1[j,h,f] = nodes[j]*w1[h,f]
// => es[h,j] = nodes[j] * (w1[h,:].asrc[h,:]), ed similarly.
// ---------------------------------------------------------------------------
__global__ void gat1_scores(const float* __restrict__ nodes, const float* __restrict__ w1,
                            const float* __restrict__ as1, const float* __restrict__ ad1,
                            float* __restrict__ es, float* __restrict__ ed) {
  const int p = blockIdx.x * blockDim.x + threadIdx.x;
  if (p >= NPIX) return;
  const float nv = nodes[p];
  for (int h = 0; h < 3; ++h) {
    float cs = 0.f, cd = 0.f;
#pragma unroll
    for (int f = 0; f < 8; ++f) { cs += w1[h * 8 + f] * as1[h * 8 + f]; cd += w1[h * 8 + f] * ad1[h * 8 + f]; }
    es[h * NPIX + p] = nv * cs;
    ed[h * NPIX + p] = nv * cd;
  }
}

// ---------------------------------------------------------------------------
// GAT layer 1 aggregation: masked softmax over sources reduces to a weighted
// scalar sum of nodes[] per (head, dst). One block per destination pixel.
// ---------------------------------------------------------------------------
__global__ void gat1_agg(const float* __restrict__ es, const float* __restrict__ ed,
                         const float* __restrict__ nodes, const float* __restrict__ mask,
                         const float* __restrict__ w1, const float* __restrict__ bias1,
                         float* __restrict__ g1) {
  __shared__ float red[256];
  __shared__ float red2[256];
  __shared__ float wsum[3];
  const int i = blockIdx.x, tid = threadIdx.x;
  for (int h = 0; h < 3; ++h) {
    const float edi = ed[h * NPIX + i];
    float lmax = -1e30f;
    for (int j = tid; j < NPIX; j += 256)
      if (mask[j] > 0.5f || j == i) lmax = fmaxf(lmax, lrelu02(edi + es[h * NPIX + j]));
    red[tid] = lmax; __syncthreads();
    for (int st = 128; st > 0; st >>= 1) { if (tid < st) red[tid] = fmaxf(red[tid], red[tid + st]); __syncthreads(); }
    const float mx = red[0]; __syncthreads();
    float lden = 0.f, lnum = 0.f;
    for (int j = tid; j < NPIX; j += 256)
      if (mask[j] > 0.5f || j == i) {
        float ex = __expf(lrelu02(edi + es[h * NPIX + j]) - mx);
        lden += ex; lnum += ex * nodes[j];
      }
    red[tid] = lden; red2[tid] = lnum; __syncthreads();
    for (int st = 128; st > 0; st >>= 1) {
      if (tid < st) { red[tid] += red[tid + st]; red2[tid] += red2[tid + st]; }
      __syncthreads();
    }
    if (tid == 0) wsum[h] = red2[0] / red[0];
    __syncthreads();
  }
  if (tid < 24) {
    float r = wsum[tid >> 3] * w1[tid] + bias1[tid];
    g1[(size_t)i * 24 + tid] = r > 0.f ? r : 0.f;
  }
}

// z2[j,25] = g1[j,:24] @ w_gat2[24,25]; es2/ed2 = z2 . asrc2/adst2
__global__ void gat2_feats(const float* __restrict__ g1, const float* __restrict__ w2,
                           const float* __restrict__ as2, const float* __restrict__ ad2,
                           float* __restrict__ z2, float* __restrict__ es2, float* __restrict__ ed2) {
  const int j = blockIdx.x * blockDim.x + threadIdx.x;
  if (j >= NPIX) return;
  float zs[25];
#pragma unroll
  for (int c = 0; c < 25; ++c) zs[c] = 0.f;
  for (int k = 0; k < 24; ++k) {
    const float gv = g1[(size_t)j * 24 + k];
#pragma unroll
    for (int c = 0; c < 25; ++c) zs[c] += gv * w2[k * 25 + c];
  }
  float cs = 0.f, cd = 0.f;
#pragma unroll
  for (int c = 0; c < 25; ++c) {
    z2[(size_t)j * 25 + c] = zs[c];
    cs += zs[c] * as2[c]; cd += zs[c] * ad2[c];
  }
  es2[j] = cs; ed2[j] = cd;
}

// GAT layer 2 aggregation: one block per destination pixel; 25-wide weighted
// feature accumulation. g2 stored flat [j*25+c] (torch-reshape semantics).
__global__ void gat2_agg(const float* __restrict__ es2, const float* __restrict__ ed2,
                         const float* __restrict__ z2, const float* __restrict__ mask,
                         const float* __restrict__ bias2, float* __restrict__ g2) {
  __shared__ float red[256];
  __shared__ float res[26];
  const int i = blockIdx.x, tid = threadIdx.x;
  const float edi = ed2[i];
  float lmax = -1e30f;
  for (int j = tid; j < NPIX; j += 256)
    if (mask[j] > 0.5f || j == i) lmax = fmaxf(lmax, lrelu02(edi + es2[j]));
  red[tid] = lmax; __syncthreads();
  for (int st = 128; st > 0; st >>= 1) { if (tid < st) red[tid] = fmaxf(red[tid], red[tid + st]); __syncthreads(); }
  const float mx = red[0]; __syncthreads();
  float lden = 0.f;
  float lacc[25];
#pragma unroll
  for (int c = 0; c < 25; ++c) lacc[c] = 0.f;
  for (int j = tid; j < NPIX; j += 256)
    if (mask[j] > 0.5f || j == i) {
      const float ex = __expf(lrelu02(edi + es2[j]) - mx);
      lden += ex;
      const float* zp = z2 + (size_t)j * 25;
#pragma unroll
      for (int c = 0; c < 25; ++c) lacc[c] += ex * zp[c];
    }
  red[tid] = lden; __syncthreads();
  for (int st = 128; st > 0; st >>= 1) { if (tid < st) red[tid] += red[tid + st]; __syncthreads(); }
  if (tid == 0) res[25] = red[0];
  __syncthreads();
  for (int c = 0; c < 25; ++c) {
    red[tid] = lacc[c]; __syncthreads();
    for (int st = 128; st > 0; st >>= 1) { if (tid < st) red[tid] += red[tid + st]; __syncthreads(); }
    if (tid == 0) res[c] = red[0];
    __syncthreads();
  }
  if (tid < 25) g2[(size_t)i * 25 + tid] = res[tid] / res[25] + bias2[tid];
}

// ---------------------------------------------------------------------------
// Host-side orchestration
// ---------------------------------------------------------------------------
extern "C" void kernel_launch(void* const* d_in, const int* in_sizes, int n_in,
                              void* d_out, int out_size, void* d_ws, size_t ws_size,
                              hipStream_t stream) {
  (void)in_sizes; (void)n_in; (void)out_size; (void)ws_size;
  const int N = NPIX;
  // Input order (setup_inputs dict order): x, then params in insertion order.
  const float* x       = (const float*)d_in[0];
  const float* g1p     = (const float*)d_in[1];
  const float* bt1     = (const float*)d_in[2];
  const float* w_con1  = (const float*)d_in[3];
  const float* b_con1  = (const float*)d_in[4];
  const float* w_cond1 = (const float*)d_in[5];
  const float* b_cond1 = (const float*)d_in[6];
  const float* g2p     = (const float*)d_in[7];
  const float* bt2     = (const float*)d_in[8];
  const float* w_con2  = (const float*)d_in[9];
  const float* b_con2  = (const float*)d_in[10];
  const float* w_cond2 = (const float*)d_in[11];
  const float* b_cond2 = (const float*)d_in[12];
  const float* g4p     = (const float*)d_in[13];
  const float* bt4     = (const float*)d_in[14];
  const float* w_mask  = (const float*)d_in[15];
  const float* b_mask  = (const float*)d_in[16];
  const float* w_trans = (const float*)d_in[17];
  const float* b_trans = (const float*)d_in[18];
  const float* w_gat1  = (const float*)d_in[19];
  const float* asrc1   = (const float*)d_in[20];
  const float* adst1   = (const float*)d_in[21];
  const float* bias1   = (const float*)d_in[22];
  const float* w_gat2  = (const float*)d_in[23];
  const float* asrc2   = (const float*)d_in[24];
  const float* adst2   = (const float*)d_in[25];
  const float* bias2   = (const float*)d_in[26];
  const float* gcat    = (const float*)d_in[27];
  const float* btcat   = (const float*)d_in[28];
  const float* w_con4  = (const float*)d_in[29];
  const float* b_con4  = (const float*)d_in[30];
  const float* w_cond4 = (const float*)d_in[31];
  const float* b_cond4 = (const float*)d_in[32];
  const float* g5p     = (const float*)d_in[33];
  const float* bt5     = (const float*)d_in[34];
  const float* w_con5  = (const float*)d_in[35];
  const float* b_con5  = (const float*)d_in[36];
  const float* w_cond5 = (const float*)d_in[37];
  const float* b_cond5 = (const float*)d_in[38];
  const float* w_con6  = (const float*)d_in[39];
  const float* b_con6  = (const float*)d_in[40];

  // Workspace layout (floats)
  float* W  = (float*)d_ws;
  float* s_ = W;           // 320
  float* t_ = W + 320;     // 320
  float* Ap = W + 640;     // 112*304 folded weights
  float* bp_ = Ap + 112 * 304;  // 128 folded bias
  size_t off = 640 + 112 * 304 + 128;
  float* out1   = W + off; off += (size_t)100 * N;
  float* buf300 = W + off; off += (size_t)304 * N;   // cat2 = [cond1(100); x(200)] + 4 pad rows
  float* out2a  = W + off; off += (size_t)100 * N;
  float* out2   = W + off; off += (size_t)25 * N;    // kept for cat5/cat6
  float* xx     = W + off; off += (size_t)25 * N;
  float* maskb  = W + off; off += (size_t)N;
  float* nodes  = W + off; off += (size_t)N;
  float* es1    = W + off; off += (size_t)3 * N;
  float* ed1    = W + off; off += (size_t)3 * N;
  float* g1b    = W + off; off += (size_t)24 * N;
  float* z2b    = W + off; off += (size_t)25 * N;
  float* es2    = W + off; off += (size_t)N;
  float* ed2    = W + off; off += (size_t)N;
  float* g2b    = W + off; off += (size_t)25 * N;
  float* buf50  = W + off; off += (size_t)56 * N;    // cat4 + 6 pad rows
  float* out4a  = W + off; off += (size_t)100 * N;
  float* buf125 = W + off; off += (size_t)128 * N;   // cat5/cat6 + 3 pad rows
  float* out5a  = W + off; off += (size_t)100 * N;

  // Zero the K-pad rows of the GEMM input slabs (deterministic every call;
  // padded A' columns are zero so these only need to be finite).
  hipMemsetAsync(buf300 + (size_t)300 * N, 0, (size_t)4 * N * sizeof(float), stream);
  hipMemsetAsync(buf50  + (size_t)50  * N, 0, (size_t)6 * N * sizeof(float), stream);
  hipMemsetAsync(buf125 + (size_t)125 * N, 0, (size_t)3 * N * sizeof(float), stream);

  const dim3 gemmBlk(128);

  // --- stage 1: bn1 + conv1 (1x1 200->100) + sigmoid ---
  bn_stats_fold<<<200, 256, 0, stream>>>(x, g1p, bt1, s_, t_);
  fold_weights<<<112, 256, 0, stream>>>(w_con1, b_con1, s_, t_, Ap, bp_, 100, 200, 200);
  gemm_wmma_f32<<<dim3(7, N / 64), gemmBlk, 0, stream>>>(Ap, x, bp_, out1, 100, 200, N, 1);
  grouped_conv5x5_sig<<<(100 * N + 255) / 256, 256, 0, stream>>>(out1, w_cond1, b_cond1, buf300, 100, 1);
  hipMemcpyAsync(buf300 + (size_t)100 * N, x, (size_t)200 * N * sizeof(float),
                 hipMemcpyDeviceToDevice, stream);

  // --- stage 2: bn2 + conv2 (1x1 300->100) + sigmoid; cond2 grouped (25x4) ---
  bn_stats_fold<<<300, 256, 0, stream>>>(buf300, g2p, bt2, s_, t_);
  fold_weights<<<112, 256, 0, stream>>>(w_con2, b_con2, s_, t_, Ap, bp_, 100, 300, 304);
  gemm_wmma_f32<<<dim3(7, N / 64), gemmBlk, 0, stream>>>(Ap, buf300, bp_, out2a, 100, 304, N, 1);
  grouped_conv5x5_sig<<<(25 * N + 255) / 256, 256, 0, stream>>>(out2a, w_cond2, b_cond2, out2, 25, 4);

  // --- bn4 -> xx (materialized) ---
  bn_stats_fold<<<25, 256, 0, stream>>>(out2, g4p, bt4, s_, t_);
  bn_apply<<<(25 * N + 255) / 256, 256, 0, stream>>>(out2, s_, t_, xx, 25 * N);

  // --- attention mask + node features ---
  att_mask_nodes<<<1, 1024, 0, stream>>>(xx, w_mask, b_mask, w_trans, b_trans, maskb, nodes);

  // --- GAT layer 1 ---
  gat1_scores<<<N / 256, 256, 0, stream>>>(nodes, w_gat1, asrc1, adst1, es1, ed1);
  gat1_agg<<<N, 256, 0, stream>>>(es1, ed1, nodes, maskb, w_gat1, bias1, g1b);

  // --- GAT layer 2 ---
  gat2_feats<<<N / 256, 256, 0, stream>>>(g1b, w_gat2, asrc2, adst2, z2b, es2, ed2);
  gat2_agg<<<N, 256, 0, stream>>>(es2, ed2, z2b, maskb, bias2, g2b);

  // --- cat4 = [xx ; gcn(flat reinterpret of g2)] ---
  hipMemcpyAsync(buf50, xx, (size_t)25 * N * sizeof(float), hipMemcpyDeviceToDevice, stream);
  hipMemcpyAsync(buf50 + (size_t)25 * N, g2b, (size_t)25 * N * sizeof(float),
                 hipMemcpyDeviceToDevice, stream);

  // --- stage 4: bncat + conv4 (1x1 50->100) + sigmoid; depthwise cond4 ---
  bn_stats_fold<<<50, 256, 0, stream>>>(buf50, gcat, btcat, s_, t_);
  fold_weights<<<112, 256, 0, stream>>>(w_con4, b_con4, s_, t_, Ap, bp_, 100, 50, 56);
  gemm_wmma_f32<<<dim3(7, N / 64), gemmBlk, 0, stream>>>(Ap, buf50, bp_, out4a, 100, 56, N, 1);
  grouped_conv5x5_sig<<<(100 * N + 255) / 256, 256, 0, stream>>>(out4a, w_cond4, b_cond4, buf125, 100, 1);
  hipMemcpyAsync(buf125 + (size_t)100 * N, out2, (size_t)25 * N * sizeof(float),
                 hipMemcpyDeviceToDevice, stream);

  // --- stage 5: bn5 + conv5 (1x1 125->100) + sigmoid; depthwise cond5 ---
  bn_stats_fold<<<125, 256, 0, stream>>>(buf125, g5p, bt5, s_, t_);
  fold_weights<<<112, 256, 0, stream>>>(w_con5, b_con5, s_, t_, Ap, bp_, 100, 125, 128);
  gemm_wmma_f32<<<dim3(7, N / 64), gemmBlk, 0, stream>>>(Ap, buf125, bp_, out5a, 100, 128, N, 1);
  // overwrite rows 0..99 with out5; rows 100..124 still hold out2 -> cat6 in place
  grouped_conv5x5_sig<<<(100 * N + 255) / 256, 256, 0, stream>>>(out5a, w_cond5, b_cond5, buf125, 100, 1);

  // --- conv6 (1x1 125->17), no bn, no activation -> d_out [17,64,64] ---
  fold_weights<<<32, 256, 0, stream>>>(w_con6, b_con6, nullptr, nullptr, Ap, bp_, 17, 125, 128);
  gemm_wmma_f32<<<dim3(2, N / 64), gemmBlk, 0, stream>>>(Ap, buf125, bp_, (float*)d_out, 17, 128, N, 0);
}